// OtsuPtPIL_1503238553780
// MI455X (gfx1250) — compile-verified
//
#include <hip/hip_runtime.h>
#include <hip/hip_bf16.h>

typedef __attribute__((ext_vector_type(2))) float v2f;
typedef __attribute__((ext_vector_type(8))) float v8f;

#define NBINS 256
#define IMG_B 64
#define IMG_HW (512 * 512)
#define CPB 64                 // histogram blocks per image
// pixels per hist block = IMG_HW / CPB = 4096 = 256 threads * 4 float4

__device__ __forceinline__ float gray_of(float r, float g, float b) {
    float y = 0.299f * r + 0.587f * g + 0.114f * b;
    return fminf(fmaxf(y, 0.0f), 1.0f);
}

__global__ void zero_hist(unsigned int* __restrict__ hist) {
    int i = blockIdx.x * blockDim.x + threadIdx.x;
    if (i < IMG_B * NBINS) hist[i] = 0u;
}

__device__ __forceinline__ void hist_px(float r, float g, float b, unsigned int* lh) {
    float y = gray_of(r, g, b);
    int q = (int)fminf(y * 255.0f, 255.0f);   // y in [0,1] -> trunc == floor
    atomicAdd(&lh[q], 1u);
}

__global__ void hist_kernel(const float* __restrict__ x, unsigned int* __restrict__ hist) {
    __shared__ unsigned int lh[NBINS];
    int tid = threadIdx.x;                    // 256 threads
    lh[tid] = 0u;
    __syncthreads();

    int img = blockIdx.x / CPB;
    int chunk = blockIdx.x % CPB;
    const float4* r4 = (const float4*)(x + (size_t)img * 3 * IMG_HW);
    const float4* g4 = r4 + (IMG_HW / 4);
    const float4* b4 = g4 + (IMG_HW / 4);
    int base4 = chunk * 1024;                 // 1024 float4 per chunk
#pragma unroll
    for (int i = 0; i < 4; ++i) {
        int idx4 = base4 + i * 256 + tid;     // coalesced b128 loads
        float4 r = r4[idx4];
        float4 g = g4[idx4];
        float4 b = b4[idx4];
        hist_px(r.x, g.x, b.x, lh);
        hist_px(r.y, g.y, b.y, lh);
        hist_px(r.z, g.z, b.z, lh);
        hist_px(r.w, g.w, b.w, lh);
    }
    __syncthreads();
    atomicAdd(&hist[img * NBINS + tid], lh[tid]);
}

// One wave32 per image. Cumulative sums done as triangular matmuls on the
// WMMA unit: hist reshaped 16x16 (segment s = row, j = K), P = M x U with
// U[k][j] = (k <= j) built in-register, chained over four K=4 WMMA steps.
__global__ void otsu_kernel(const unsigned int* __restrict__ hist, float* __restrict__ thr) {
    __shared__ float h[NBINS];
    __shared__ float ps[NBINS];   // within-segment inclusive cumsum of counts
    __shared__ float pw[NBINS];   // same for counts * bin
    __shared__ float eo[16];      // exclusive segment offsets (counts)
    __shared__ float ew[16];      // exclusive segment offsets (weighted)
    __shared__ float rv[32];
    __shared__ int   ri[32];

    int lane = threadIdx.x;       // 0..31, full wave, EXEC all ones
    int img = blockIdx.x;
    const unsigned int* hp = hist + img * NBINS;
#pragma unroll
    for (int k = 0; k < 8; ++k) h[k * 32 + lane] = (float)hp[k * 32 + lane];
    __syncthreads();

    int row  = lane & 15;         // A-matrix M row held by this lane
    int half = lane >> 4;         // 0: K pair {0,1}; 1: K pair {2,3}
    int base = half * 2;
    int col  = lane & 15;         // B/D column N

    v8f acc  = {0.f, 0.f, 0.f, 0.f, 0.f, 0.f, 0.f, 0.f};
    v8f accw = {0.f, 0.f, 0.f, 0.f, 0.f, 0.f, 0.f, 0.f};
#pragma unroll
    for (int kb = 0; kb < 4; ++kb) {
        int k0 = 4 * kb + base;
        v2f a, aw, bb;
        a.x  = h[16 * row + k0];
        a.y  = h[16 * row + k0 + 1];
        aw.x = a.x * (float)(16 * row + k0);
        aw.y = a.y * (float)(16 * row + k0 + 1);
        bb.x = (k0     <= col) ? 1.0f : 0.0f;   // upper-triangular ones
        bb.y = (k0 + 1 <= col) ? 1.0f : 0.0f;
        acc  = __builtin_amdgcn_wmma_f32_16x16x4_f32(false, a,  false, bb, (short)0, acc,  false, false);
        accw = __builtin_amdgcn_wmma_f32_16x16x4_f32(false, aw, false, bb, (short)0, accw, false, false);
    }

    // D layout: VGPR r, lanes 0-15 -> M=r, lanes 16-31 -> M=r+8; N = lane&15
#pragma unroll
    for (int r = 0; r < 8; ++r) {
        int M = r + 8 * half;
        ps[16 * M + col] = acc[r];
        pw[16 * M + col] = accw[r];
    }
    __syncthreads();

    if (lane < 16) {              // exclusive prefix over 16 segment totals
        float s = 0.f, sw = 0.f;
        for (int t = 0; t < lane; ++t) {
            s  += ps[16 * t + 15];
            sw += pw[16 * t + 15];
        }
        eo[lane] = s;
        ew[lane] = sw;
    }
    __syncthreads();

    float total = ps[255] + eo[15];
    float totm  = pw[255] + ew[15];

    float best = -1.0f;           // var12 >= 0, so first candidate always wins
    int bidx = 0;
    for (int t = lane; t < NBINS - 1; t += 32) {
        int t1 = t + 1;
        float w1  = ps[t]  + eo[t  >> 4];
        float m1  = pw[t]  + ew[t  >> 4];
        float w1b = ps[t1] + eo[t1 >> 4];
        float m1b = pw[t1] + ew[t1 >> 4];
        float cnt1 = h[t1];
        float w2 = total - w1b + cnt1;
        float m2 = totm  - m1b + cnt1 * (float)t1;
        float mean1 = (w1 > 0.f) ? m1 / fmaxf(w1, 1.f) : 0.f;
        float mean2 = (w2 > 0.f) ? m2 / fmaxf(w2, 1.f) : 0.f;
        float d = mean1 - mean2;
        float v = w1 * w2 * d * d;
        if (v > best) { best = v; bidx = t; }   // ascending t: first max kept
    }
    rv[lane] = best;
    ri[lane] = bidx;
    __syncthreads();
    if (lane == 0) {
        float bv = rv[0];
        int bi = ri[0];
        for (int l = 1; l < 32; ++l) {
            if (rv[l] > bv || (rv[l] == bv && ri[l] < bi)) { bv = rv[l]; bi = ri[l]; }
        }
        thr[img] = (float)bi / 255.0f;
    }
}

__global__ void thresh_kernel(const float* __restrict__ x, const float* __restrict__ thr,
                              float* __restrict__ out) {
    int gid = blockIdx.x * blockDim.x + threadIdx.x;   // float4 index over B*HW/4
    int img = gid / (IMG_HW / 4);
    int p4  = gid % (IMG_HW / 4);
    float t = thr[img];                                // uniform within block
    const float4* r4 = (const float4*)(x + (size_t)img * 3 * IMG_HW);
    const float4* g4 = r4 + (IMG_HW / 4);
    const float4* b4 = g4 + (IMG_HW / 4);
    float4 r = r4[p4];
    float4 g = g4[p4];
    float4 b = b4[p4];
    float4 o;
    o.x = (gray_of(r.x, g.x, b.x) > t) ? 1.0f : 0.0f;
    o.y = (gray_of(r.y, g.y, b.y) > t) ? 1.0f : 0.0f;
    o.z = (gray_of(r.z, g.z, b.z) > t) ? 1.0f : 0.0f;
    o.w = (gray_of(r.w, g.w, b.w) > t) ? 1.0f : 0.0f;
    ((float4*)out)[gid] = o;
}

extern "C" void kernel_launch(void* const* d_in, const int* in_sizes, int n_in,
                              void* d_out, int out_size, void* d_ws, size_t ws_size,
                              hipStream_t stream) {
    (void)in_sizes; (void)n_in; (void)out_size; (void)ws_size;
    const float* x = (const float*)d_in[0];
    float* out = (float*)d_out;

    unsigned int* hist = (unsigned int*)d_ws;                       // 64 KiB
    float* thr = (float*)((char*)d_ws + IMG_B * NBINS * sizeof(unsigned int)); // 256 B

    zero_hist<<<(IMG_B * NBINS + 255) / 256, 256, 0, stream>>>(hist);
    hist_kernel<<<IMG_B * CPB, 256, 0, stream>>>(x, hist);
    otsu_kernel<<<IMG_B, 32, 0, stream>>>(hist, thr);
    thresh_kernel<<<(IMG_B * (IMG_HW / 4)) / 256, 256, 0, stream>>>(x, thr, out);
}